// find_k_nearest_neighbors_23192823398853
// MI455X (gfx1250) — compile-verified
//
#include <hip/hip_runtime.h>
#include <cstdint>

#define N_OBS   2048
#define M_PAN   40000
#define K_NN    16
#define L_DIM   24
#define BC_DIM  32                       // B*C = 8*4
#define NCHUNK  8
#define CHUNK   5000                     // M_PAN / NCHUNK
#define TILES   ((CHUNK + 15) / 16)      // 313 (last tile has 8 valid cols)
#define ROWG    (N_OBS / 16)             // 128

typedef __attribute__((ext_vector_type(2))) float        v2f;
typedef __attribute__((ext_vector_type(8))) float        v8f;
typedef __attribute__((ext_vector_type(4))) unsigned int v4u;
typedef __attribute__((ext_vector_type(8))) int          v8i;
typedef __attribute__((ext_vector_type(4))) int          v4i;

#if defined(__has_builtin)
#if __has_builtin(__builtin_amdgcn_tensor_load_to_lds) && __has_builtin(__builtin_amdgcn_s_wait_tensorcnt)
#define HAVE_TDM 1
#endif
#endif

// Branchless sorted-insert into a 16-deep register list (cndmask per level,
// constant indices only -> arrays stay in VGPRs). Call only under a guard
// (cd < DARR[15]) so the common path is a single compare.
#define INSERT16(DARR, IARR, CD, CI)                                    \
    do {                                                                \
        float _cd = (CD); int _ci = (CI);                               \
        _Pragma("unroll")                                               \
        for (int _q = 0; _q < K_NN; ++_q) {                             \
            const bool  _sw = _cd < DARR[_q];                           \
            const float _nd = _sw ? _cd : DARR[_q];                     \
            const int   _ni = _sw ? _ci : IARR[_q];                     \
            _cd = _sw ? DARR[_q] : _cd;                                 \
            _ci = _sw ? IARR[_q] : _ci;                                 \
            DARR[_q] = _nd; IARR[_q] = _ni;                             \
        }                                                               \
    } while (0)

// ---------------------------------------------------------------------------
// Kernel 1: per-(rowgroup, M-chunk) wave computes d2 tiles via WMMA and keeps
// a register-resident sorted top-16 per row. Writes partial lists to ws.
// ---------------------------------------------------------------------------
__global__ __launch_bounds__(32)
void knn_select_kernel(const float2* __restrict__ cobs2,
                       const float2* __restrict__ cpan2,
                       float* __restrict__ wsD, int* __restrict__ wsI) {
    __shared__ float2 s_chunk[CHUNK];   // 40 KB staged cpan chunk (reused for merge)
    __shared__ float  s_tile[256];      // one 16x16 d2 tile

    const int lane = threadIdx.x;
    const int rg = blockIdx.x & (ROWG - 1);
    const int ch = blockIdx.x >> 7;
    const int r0 = rg * 16;
    const int hi = lane >> 4;           // 0: K0/K1 + rows 0-7 slots, 1: K2/K3 + rows 8-15

    // A matrix: lanes 0-15 hold row M=lane, VGPR0=K0, VGPR1=K1. Scale by -2 so
    // D = A*B + C = (na+nb) - 2*dot = d2 directly. Lanes 16-31 (K2,K3) are zero.
    v2f a = {0.0f, 0.0f};
    float naR = 0.0f;
    if (lane < 16) {
        float2 cc = cobs2[r0 + lane];
        a.x = -2.0f * cc.x;
        a.y = -2.0f * cc.y;
        naR = cc.x * cc.x + cc.y * cc.y;
    }
    float naL[8];
    #pragma unroll
    for (int v = 0; v < 8; ++v)          // C/D row of (v,lane) is v + 8*hi
        naL[v] = __shfl(naR, v + 8 * hi, 32);

    // Stage this chunk of cpan into LDS via the Tensor Data Mover.
#if HAVE_TDM
    {
        unsigned long long ga = (unsigned long long)(uintptr_t)(cpan2 + (size_t)ch * CHUNK);
        unsigned int la = (unsigned int)(uintptr_t)&s_chunk[0];
        const unsigned tdim0 = M_PAN * 2u;   // whole cpan in dwords
        const unsigned tile0 = CHUNK * 2u;   // chunk in dwords
        v4u g0;
        g0[0] = 1u;                                          // count=1, user descriptor
        g0[1] = la;                                          // lds_addr (bytes)
        g0[2] = (unsigned int)(ga & 0xffffffffull);          // global_addr[31:0]
        g0[3] = (unsigned int)((ga >> 32) & 0x1ffffffull) | 0x80000000u; // addr[56:32] | type=2
        v8i g1;
        g1[0] = (int)(2u << 16);                             // data_size = 4 bytes
        g1[1] = (int)((tdim0 & 0xffffu) << 16);              // tensor_dim0[15:0]
        g1[2] = (int)(((tdim0 >> 16) & 0xffffu) | (1u << 16)); // tensor_dim0[31:16] | tensor_dim1=1
        g1[3] = (int)(tile0 << 16);                          // tile_dim0
        g1[4] = 1;                                           // tile_dim1 = 1 (single row)
        g1[5] = (int)tdim0;                                  // tensor_dim0_stride lo
        g1[6] = 0;
        g1[7] = 0;
        v4i g2 = {0, 0, 0, 0};
        v4i g3 = {0, 0, 0, 0};
        v8i g4 = {0, 0, 0, 0, 0, 0, 0, 0};                   // trailing group (clang-23 form)
        __builtin_amdgcn_tensor_load_to_lds(g0, g1, g2, g3, g4, 0);
        __builtin_amdgcn_s_wait_tensorcnt(0);
    }
#else
    for (int i = lane; i < CHUNK; i += 32)
        s_chunk[i] = cpan2[(size_t)ch * CHUNK + i];
#endif
    __syncthreads();

    // Per-lane sorted top-16: lane r and lane r+16 both own row r (8 cols each).
    float dist[K_NN];
    int   idxl[K_NN];
    #pragma unroll
    for (int q = 0; q < K_NN; ++q) { dist[q] = 3.4e38f; idxl[q] = 0; }

    const int row   = lane & 15;
    const int jbase = hi * 8;
    const int gbase = ch * CHUNK;

    #pragma unroll 1
    for (int t = 0; t < TILES; ++t) {
        const int cbase = t * 16;
        // B matrix: lanes 0-15 hold col N=lane, VGPR0=K0 row (x), VGPR1=K1 row (y).
        v2f bm = {0.0f, 0.0f};
        float nb = 1.0e30f;              // invalid cols get a huge norm -> huge d2
        if (lane < 16) {
            int ci = cbase + lane;
            if (ci < CHUNK) {
                float2 pp = s_chunk[ci];
                bm.x = pp.x; bm.y = pp.y;
                nb = pp.x * pp.x + pp.y * pp.y;
            }
        }
        float nbB = __shfl(nb, row, 32); // broadcast col-norm of col (lane&15)

        v8f cmat;
        #pragma unroll
        for (int v = 0; v < 8; ++v) cmat[v] = naL[v] + nbB;

        v8f dmat = __builtin_amdgcn_wmma_f32_16x16x4_f32(
            false, a, false, bm, (short)0, cmat, false, false);

        // Redistribute the 16x16 tile through LDS so each lane scans one row.
        #pragma unroll
        for (int v = 0; v < 8; ++v)
            s_tile[(v + 8 * hi) * 16 + row] = dmat[v];
        __syncthreads();

        #pragma unroll 1                 // keep ONE insert blob, not eight
        for (int jj = 0; jj < 8; ++jj) {
            const int j = jbase + jj;
            const float dv = s_tile[row * 16 + j];
            if (dv < dist[K_NN - 1]) {
                INSERT16(dist, idxl, dv, gbase + cbase + j);
            }
        }
        __syncthreads();
    }

    // Merge partner lane (r+16) list into lane r's list via LDS (s_chunk reuse).
    float* pD = (float*)s_chunk;
    int*   pI = (int*)((float*)s_chunk + 256);
    if (hi) {
        #pragma unroll
        for (int q = 0; q < K_NN; ++q) {
            pD[row * K_NN + q] = dist[q];
            pI[row * K_NN + q] = idxl[q];
        }
    }
    __syncthreads();
    if (lane < 16) {
        #pragma unroll 1
        for (int q = 0; q < K_NN; ++q) {
            const float pd = pD[row * K_NN + q];
            const int   pi = pI[row * K_NN + q];
            if (pd < dist[K_NN - 1]) {
                INSERT16(dist, idxl, pd, pi);
            }
        }
        const size_t base = ((size_t)ch * N_OBS + (r0 + lane)) * K_NN;
        #pragma unroll
        for (int kk = 0; kk < K_NN; ++kk) {
            wsD[base + kk] = dist[kk];
            wsI[base + kk] = idxl[kk];
        }
    }
}

// ---------------------------------------------------------------------------
// Kernel 2: one thread per query row merges 8 sorted partial lists -> final 16.
// Also emits cpan_k and the final index table for the gather.
// ---------------------------------------------------------------------------
__global__ __launch_bounds__(256)
void knn_merge_kernel(const float* __restrict__ wsD, const int* __restrict__ wsI,
                      const float2* __restrict__ cpan2,
                      int* __restrict__ wsF, float* __restrict__ outC) {
    const int n = blockIdx.x * 256 + threadIdx.x;
    if (n >= N_OBS) return;

    float dist[K_NN];
    int   idxl[K_NN];
    #pragma unroll
    for (int q = 0; q < K_NN; ++q) { dist[q] = 3.4e38f; idxl[q] = 0; }

    // Candidates iterated in chunk-major (= ascending panel index) order so
    // equal distances keep the lower index first (matches lax.top_k).
    #pragma unroll 1
    for (int c = 0; c < NCHUNK * K_NN; ++c) {
        const int ch = c >> 4;
        const int kk = c & (K_NN - 1);
        const size_t base = ((size_t)ch * N_OBS + n) * K_NN;
        const float d  = wsD[base + kk];
        const int   id = wsI[base + kk];
        if (d < dist[K_NN - 1]) {
            INSERT16(dist, idxl, d, id);
        }
    }

    #pragma unroll
    for (int kk = 0; kk < K_NN; ++kk) {
        const int id = idxl[kk];
        wsF[n * K_NN + kk] = id;
        const float2 cc = cpan2[id];
        outC[((size_t)n * K_NN + kk) * 2 + 0] = cc.x;
        outC[((size_t)n * K_NN + kk) * 2 + 1] = cc.y;
    }
}

// ---------------------------------------------------------------------------
// Kernel 3: bandwidth-bound gather. One thread per float4 (24 floats = 6 vec4
// per (b,c,n,k) row; rows are 96 B so vec4 accesses stay 16 B aligned).
// ---------------------------------------------------------------------------
__global__ __launch_bounds__(256)
void knn_gather_kernel(const float4* __restrict__ pan4,
                       const int* __restrict__ wsF,
                       float4* __restrict__ out4) {
    unsigned tid  = blockIdx.x * 256u + threadIdx.x;
    unsigned q    = tid % 6u;
    unsigned rest = tid / 6u;
    unsigned kk   = rest & 15u;  rest >>= 4;
    unsigned n    = rest & 2047u; rest >>= 11;
    unsigned bc   = rest;                       // 0..31
    int idx = wsF[n * K_NN + kk];
    out4[tid] = pan4[((size_t)bc * M_PAN + (unsigned)idx) * 6u + q];
}

// ---------------------------------------------------------------------------
extern "C" void kernel_launch(void* const* d_in, const int* in_sizes, int n_in,
                              void* d_out, int out_size, void* d_ws, size_t ws_size,
                              hipStream_t stream) {
    // inputs: 0=obs_his (unused), 1=cobs, 2=pan_fut, 3=cpan, 4=k (fixed 16)
    const float2* cobs2 = (const float2*)d_in[1];
    const float*  pan   = (const float*)d_in[2];
    const float2* cpan2 = (const float2*)d_in[3];
    float* out = (float*)d_out;

    // workspace layout: partial dists (1 MB) | partial idx (1 MB) | final idx (128 KB)
    float* wsD = (float*)d_ws;
    int*   wsI = (int*)((char*)d_ws + (size_t)NCHUNK * N_OBS * K_NN * 4);
    int*   wsF = (int*)((char*)d_ws + (size_t)NCHUNK * N_OBS * K_NN * 8);

    knn_select_kernel<<<dim3(ROWG * NCHUNK), dim3(32), 0, stream>>>(cobs2, cpan2, wsD, wsI);

    float* outC = out + (size_t)BC_DIM * N_OBS * K_NN * L_DIM;   // cpan_k region
    knn_merge_kernel<<<dim3(N_OBS / 256), dim3(256), 0, stream>>>(wsD, wsI, cpan2, wsF, outC);

    const unsigned total4 = BC_DIM * N_OBS * K_NN * (L_DIM / 4); // 6,291,456
    knn_gather_kernel<<<dim3(total4 / 256), dim3(256), 0, stream>>>(
        (const float4*)pan, wsF, (float4*)out);
}